// Beta_TCVAE_25056839205857
// MI455X (gfx1250) — compile-verified
//
#include <hip/hip_runtime.h>
#include <hip/hip_bf16.h>
#include <math.h>

// ---------------------------------------------------------------------------
// Beta-TCVAE loss for MI455X (gfx1250, wave32).
//   B=256, D=12288, L=32.
// Bandwidth roofline: dominated by 3*B*D*4B = 37.7MB reads -> ~1.6us @23.3TB/s.
// The pairwise term sum_l d[i,l]*e[j,l] is a 256x256x32 f32 GEMM done with
// V_WMMA_F32_16X16X4_F32 (one wave per 16x16 tile, 8 K-steps of 4).
// ---------------------------------------------------------------------------

#define BATCH 256
#define DDIM  12288
#define LDIM  32
#define BETA_F 6.0f
#define LOG2PI_F 1.8378770664093453f

typedef __attribute__((ext_vector_type(2))) float v2f;
typedef __attribute__((ext_vector_type(8))) float v8f;

__device__ __forceinline__ float waveSum(float v) {
#pragma unroll
    for (int off = 16; off > 0; off >>= 1) v += __shfl_xor(v, off, 32);
    return v;
}
__device__ __forceinline__ float waveMax(float v) {
#pragma unroll
    for (int off = 16; off > 0; off >>= 1) v = fmaxf(v, __shfl_xor(v, off, 32));
    return v;
}

// ---------------------------------------------------------------------------
// Kernel A: log_px[i] = -0.5*(D*LOG2PI + sum_d( lv + (x-m)^2*exp(-lv) ))
// One block (256 thr = 8 waves) per batch row; float4 (b128) streaming loads.
// ---------------------------------------------------------------------------
__global__ void __launch_bounds__(256)
logpx_kernel(const float* __restrict__ target,
             const float* __restrict__ xm,
             const float* __restrict__ xlv,
             float* __restrict__ log_px) {
    const int i    = blockIdx.x;
    const int tid  = threadIdx.x;
    const int lane = tid & 31;
    const int w    = tid >> 5;

    const float4* tg = (const float4*)(target + (size_t)i * DDIM);
    const float4* mn = (const float4*)(xm     + (size_t)i * DDIM);
    const float4* lv = (const float4*)(xlv    + (size_t)i * DDIM);

    float acc = 0.0f;
    // DDIM/4 = 3072 float4 per row; 256 threads -> 12 iterations.
#pragma unroll 4
    for (int it = 0; it < 12; ++it) {
        const int idx = it * 256 + tid;
        if (it + 1 < 12) {
            __builtin_prefetch(&tg[idx + 256], 0, 3);   // global_prefetch_b8
            __builtin_prefetch(&mn[idx + 256], 0, 3);
            __builtin_prefetch(&lv[idx + 256], 0, 3);
        }
        const float4 x = tg[idx];
        const float4 m = mn[idx];
        const float4 v = lv[idx];
        float dx;
        dx = x.x - m.x; acc += v.x + dx * dx * __expf(-v.x);
        dx = x.y - m.y; acc += v.y + dx * dx * __expf(-v.y);
        dx = x.z - m.z; acc += v.z + dx * dx * __expf(-v.z);
        dx = x.w - m.w; acc += v.w + dx * dx * __expf(-v.w);
    }

    __shared__ float red[8];
    const float s = waveSum(acc);
    if (lane == 0) red[w] = s;
    __syncthreads();
    if (tid == 0) {
        float t = 0.0f;
#pragma unroll
        for (int k = 0; k < 8; ++k) t += red[k];
        log_px[i] = -0.5f * ((float)DDIM * LOG2PI_F + t);
    }
}

// ---------------------------------------------------------------------------
// Kernel B: per-(b,l) prep. One wave per batch row (L == warpSize == 32).
//   d[b,l] = (z-z_mean)^2 ; e[b,l] = exp(-z_log_var)
//   vsum[b] = sum_l z_log_var ; log_qzx[b] ; log_pz[b]
// ---------------------------------------------------------------------------
__global__ void __launch_bounds__(256)
prep_kernel(const float* __restrict__ z,
            const float* __restrict__ zm,
            const float* __restrict__ zlv,
            float* __restrict__ d_out,
            float* __restrict__ e_out,
            float* __restrict__ vsum,
            float* __restrict__ log_qzx,
            float* __restrict__ log_pz) {
    const int tid = blockIdx.x * 256 + threadIdx.x;   // 0..8191
    const int b   = tid >> 5;
    const int l   = tid & 31;

    const float zz = z[tid];
    const float m  = zm[tid];
    const float lv = zlv[tid];

    const float diff = zz - m;
    const float dd   = diff * diff;
    const float ee   = __expf(-lv);
    d_out[tid] = dd;
    e_out[tid] = ee;

    const float s_qzx = waveSum(lv + dd * ee);
    const float s_pz  = waveSum(zz * zz);
    const float s_v   = waveSum(lv);
    if (l == 0) {
        log_qzx[b] = -0.5f * ((float)LDIM * LOG2PI_F + s_qzx);
        log_pz[b]  = -0.5f * ((float)LDIM * LOG2PI_F + s_pz);
        vsum[b]    = s_v;
    }
}

// ---------------------------------------------------------------------------
// Kernel C: S[i,j] = sum_l d[i,l]*e[j,l] via V_WMMA_F32_16X16X4_F32, fused
// into rowsum[i,j] = -0.5*(L*LOG2PI + vsum[j] + S[i,j]).
// One wave (32 thr) per 16x16 tile; K=32 in 8 steps of 4.
// A-fragment layout (ISA 7.12.2, 32-bit A 16x4): lane t, m=t&15:
//   VGPR0 = A[m][k0 + 2*(t>>4)], VGPR1 = A[m][k0 + 2*(t>>4) + 1]
// B-fragment mirrors with columns striped across lanes.
// C/D layout: VGPR r -> M = r + 8*(t>>4), N = t&15.
// ---------------------------------------------------------------------------
__global__ void __launch_bounds__(32)
pairwise_wmma_kernel(const float* __restrict__ d_in,   // [B][L]
                     const float* __restrict__ e_in,   // [B][L]
                     const float* __restrict__ vsum,   // [B]
                     float* __restrict__ rowsum) {     // [B][B]
    const int i0 = blockIdx.x * 16;
    const int j0 = blockIdx.y * 16;
    const int t  = threadIdx.x;          // 0..31, full wave, EXEC all ones
    const int hl = t & 15;
    const int hi = t >> 4;               // 0 or 1

    v8f c = {};
#pragma unroll
    for (int kk = 0; kk < 8; ++kk) {
        const int k = kk * 4 + 2 * hi;
        v2f a, b;
        a.x = d_in[(i0 + hl) * LDIM + k];
        a.y = d_in[(i0 + hl) * LDIM + k + 1];
        b.x = e_in[(j0 + hl) * LDIM + k];
        b.y = e_in[(j0 + hl) * LDIM + k + 1];
        c = __builtin_amdgcn_wmma_f32_16x16x4_f32(
                /*neg_a=*/false, a, /*neg_b=*/false, b,
                /*c_mod=*/(short)0, c, /*reuse_a=*/false, /*reuse_b=*/false);
    }

    const int n  = hl;
    const float vj = vsum[j0 + n];
#pragma unroll
    for (int r = 0; r < 8; ++r) {
        const int m = r + 8 * hi;
        rowsum[(size_t)(i0 + m) * BATCH + (j0 + n)] =
            -0.5f * ((float)LDIM * LOG2PI_F + vj + c[r]);
    }
}

// ---------------------------------------------------------------------------
// Kernel D: per-i logsumexp reductions over j.
//   log_qz_prod[i] = sum_l ( lse_j( -0.5*(LOG2PI + v[j,l] + d[i,l]*e[j,l]) ) - log_nm )
//   log_qz[i]      = lse_j( rowsum[i,j] ) - log_nm
// One block (8 waves) per i; each wave handles 4 values of l, lanes stride j.
// ---------------------------------------------------------------------------
__global__ void __launch_bounds__(256)
qz_kernel(const float* __restrict__ d_in,
          const float* __restrict__ e_in,
          const float* __restrict__ zlv,
          const float* __restrict__ rowsum,
          float* __restrict__ log_qz,
          float* __restrict__ log_qz_prod) {
    const int i    = blockIdx.x;
    const int tid  = threadIdx.x;
    const int lane = tid & 31;
    const int w    = tid >> 5;

    __shared__ float part[8];
    __shared__ float lse_row_s;

    const float log_nm = __logf((float)BATCH * 202599.0f);

    float acc = 0.0f;
#pragma unroll
    for (int ll = 0; ll < 4; ++ll) {
        const int l = w * 4 + ll;
        const float dil = d_in[i * LDIM + l];
        float vals[8];
        float mx = -INFINITY;
#pragma unroll
        for (int q = 0; q < 8; ++q) {
            const int j = lane + 32 * q;
            const float val =
                -0.5f * (LOG2PI_F + zlv[j * LDIM + l] + dil * e_in[j * LDIM + l]);
            vals[q] = val;
            mx = fmaxf(mx, val);
        }
        mx = waveMax(mx);
        float s = 0.0f;
#pragma unroll
        for (int q = 0; q < 8; ++q) s += __expf(vals[q] - mx);
        s = waveSum(s);
        acc += mx + __logf(s) - log_nm;
    }
    if (lane == 0) part[w] = acc;

    if (w == 0) {
        float vals[8];
        float mx = -INFINITY;
#pragma unroll
        for (int q = 0; q < 8; ++q) {
            vals[q] = rowsum[(size_t)i * BATCH + lane + 32 * q];
            mx = fmaxf(mx, vals[q]);
        }
        mx = waveMax(mx);
        float s = 0.0f;
#pragma unroll
        for (int q = 0; q < 8; ++q) s += __expf(vals[q] - mx);
        s = waveSum(s);
        if (lane == 0) lse_row_s = mx + __logf(s);
    }
    __syncthreads();
    if (tid == 0) {
        float p = 0.0f;
#pragma unroll
        for (int k = 0; k < 8; ++k) p += part[k];
        log_qz_prod[i] = p;
        log_qz[i]      = lse_row_s - log_nm;
    }
}

// ---------------------------------------------------------------------------
// Kernel E: loss = -mean_i( log_px - log_qzx + (1-B)*log_qz
//                            + (B-1)*log_qz_prod + log_pz )
// ---------------------------------------------------------------------------
__global__ void __launch_bounds__(256)
final_kernel(const float* __restrict__ log_px,
             const float* __restrict__ log_qzx,
             const float* __restrict__ log_pz,
             const float* __restrict__ log_qz,
             const float* __restrict__ log_qz_prod,
             float* __restrict__ out) {
    const int i    = threadIdx.x;
    const int lane = i & 31;
    const int w    = i >> 5;

    const float term = log_px[i] - log_qzx[i]
                     + (1.0f - BETA_F) * log_qz[i]
                     + (BETA_F - 1.0f) * log_qz_prod[i]
                     + log_pz[i];

    __shared__ float red[8];
    const float s = waveSum(term);
    if (lane == 0) red[w] = s;
    __syncthreads();
    if (i == 0) {
        float t = 0.0f;
#pragma unroll
        for (int k = 0; k < 8; ++k) t += red[k];
        out[0] = -t / (float)BATCH;
    }
}

// ---------------------------------------------------------------------------
extern "C" void kernel_launch(void* const* d_in, const int* in_sizes, int n_in,
                              void* d_out, int out_size, void* d_ws, size_t ws_size,
                              hipStream_t stream) {
    (void)in_sizes; (void)n_in; (void)out_size; (void)ws_size;

    const float* target = (const float*)d_in[0];   // [B, D]
    const float* x_mean = (const float*)d_in[1];   // [B, D]
    const float* x_lvar = (const float*)d_in[2];   // [B, D]
    const float* z      = (const float*)d_in[3];   // [B, L]
    const float* z_mean = (const float*)d_in[4];   // [B, L]
    const float* z_lvar = (const float*)d_in[5];   // [B, L]
    float* out = (float*)d_out;

    float* ws = (float*)d_ws;
    float* w_log_px      = ws;                 // 256
    float* w_log_qzx     = ws + 256;           // 256
    float* w_log_pz      = ws + 512;           // 256
    float* w_vsum        = ws + 768;           // 256
    float* w_log_qz      = ws + 1024;          // 256
    float* w_log_qz_prod = ws + 1280;          // 256
    float* w_d           = ws + 1536;          // 8192
    float* w_e           = ws + 1536 + 8192;   // 8192
    float* w_rowsum      = ws + 1536 + 16384;  // 65536

    // A: big streaming reduction (HBM-bound part)
    logpx_kernel<<<BATCH, 256, 0, stream>>>(target, x_mean, x_lvar, w_log_px);

    // B: tiny prep (d, e, vsum, log_qzx, log_pz)
    prep_kernel<<<(BATCH * LDIM) / 256, 256, 0, stream>>>(
        z, z_mean, z_lvar, w_d, w_e, w_vsum, w_log_qzx, w_log_pz);

    // C: 256x256x32 f32 GEMM via WMMA, fused with rowsum epilogue
    pairwise_wmma_kernel<<<dim3(BATCH / 16, BATCH / 16), 32, 0, stream>>>(
        w_d, w_e, w_vsum, w_rowsum);

    // D: per-i logsumexp reductions
    qz_kernel<<<BATCH, 256, 0, stream>>>(
        w_d, w_e, z_lvar, w_rowsum, w_log_qz, w_log_qz_prod);

    // E: final scalar
    final_kernel<<<1, 256, 0, stream>>>(
        w_log_px, w_log_qzx, w_log_pz, w_log_qz, w_log_qz_prod, out);
}